// RecNet_69561290326406
// MI455X (gfx1250) — compile-verified
//
#include <hip/hip_runtime.h>

// ---------------------------------------------------------------------------
// RecNet (LIF recurrent SNN) for MI455X / gfx1250, wave32, WMMA f32_16x16x4.
//
// Per timestep t:
//   gemm = x_t @ w_in^T + z_{t-1} @ w_rec^T          (WMMA, 16x16 tile/wave)
//   v_dec = v + 0.1*(i - v);  z_t = (v_dec > 1);  v = (1-z_t)*v_dec
//   i = 0.8*i + gemm
//   y  = z_t @ w_out^T                               (tiny readout kernel)
//   lv = lv + 0.1*(li - lv);  li = 0.8*li + y;  out[t] = lv
// ---------------------------------------------------------------------------

typedef __attribute__((ext_vector_type(2))) float v2f;
typedef __attribute__((ext_vector_type(8))) float v8f;

#define T_STEPS 100
#define B_SZ    512
#define F_SZ    784
#define H_SZ    512
#define O_SZ    10

#define DV 0.1f   // DT * TAU_MEM_INV
#define DI 0.2f   // DT * TAU_SYN_INV

// ---------------------------------------------------------------------------
// Kernel 1: per-step fused GEMM (input + recurrent) + LIF state update.
// One wave per 16x16 tile of [B,H]; 32x32 = 1024 tiles; block = 4 waves.
// WMMA requires EXEC all-1s: grid is exact, no divergent early exits.
// ---------------------------------------------------------------------------
__global__ __launch_bounds__(128)
void recnet_step_kernel(const float* __restrict__ x,      // [T,B,F]
                        const float* __restrict__ w_in,   // [H,F]
                        const float* __restrict__ w_rec,  // [H,H]
                        const float* __restrict__ z_prev, // [B,H]
                        float* __restrict__ z_cur,        // [B,H]
                        float* __restrict__ v_st,         // [B,H]
                        float* __restrict__ i_st,         // [B,H]
                        int t)
{
    const int wave = blockIdx.x * (blockDim.x >> 5) + (threadIdx.x >> 5);
    const int lane = threadIdx.x & 31;
    const int half = lane >> 4;     // K-half selector for A/B operands
    const int lm   = lane & 15;     // M (for A) / N (for B,D)

    const int tile_b = (wave >> 5) << 4;   // batch-tile origin
    const int tile_h = (wave & 31) << 4;   // hidden-tile origin

    v8f acc = {};   // 16x16 f32 accumulator (D = A*B + C)

    // ---- x_t @ w_in^T : A[m,k] = x[t, tile_b+m, k], B[k,n] = w_in[tile_h+n, k]
    {
        const float* arow = x    + ((size_t)t * B_SZ + (tile_b + lm)) * F_SZ;
        const float* brow = w_in + (size_t)(tile_h + lm) * F_SZ;
        #pragma unroll 4
        for (int k = 0; k < F_SZ; k += 4) {
            v2f a = *(const v2f*)(arow + k + 2 * half);
            v2f b = *(const v2f*)(brow + k + 2 * half);
            acc = __builtin_amdgcn_wmma_f32_16x16x4_f32(
                false, a, false, b, (short)0, acc, false, false);
        }
    }

    // ---- z_{t-1} @ w_rec^T : A[m,k] = z_prev[tile_b+m, k], B[k,n] = w_rec[tile_h+n, k]
    {
        const float* arow = z_prev + (size_t)(tile_b + lm) * H_SZ;
        const float* brow = w_rec  + (size_t)(tile_h + lm) * H_SZ;
        #pragma unroll 4
        for (int k = 0; k < H_SZ; k += 4) {
            v2f a = *(const v2f*)(arow + k + 2 * half);
            v2f b = *(const v2f*)(brow + k + 2 * half);
            acc = __builtin_amdgcn_wmma_f32_16x16x4_f32(
                false, a, false, b, (short)0, acc, false, false);
        }
    }

    // ---- Elementwise LIF update. D layout: VGPR r, lane -> (M = r + 8*half, N = lm)
    const int h = tile_h + lm;
    #pragma unroll
    for (int r = 0; r < 8; ++r) {
        const int    b   = tile_b + r + 8 * half;
        const size_t idx = (size_t)b * H_SZ + h;
        const float  vo  = v_st[idx];
        const float  io  = i_st[idx];
        const float  vdec = vo + DV * (io - vo);
        const float  z    = (vdec - 1.0f) > 0.0f ? 1.0f : 0.0f;
        v_st[idx]  = (1.0f - z) * vdec;
        i_st[idx]  = io - DI * io + acc[r];
        z_cur[idx] = z;
    }
}

// ---------------------------------------------------------------------------
// Kernel 2: readout y = z_t @ w_out^T and LI filter; writes out[t].
// One thread per (b,o): 5120 threads, K=512 dot products (w_out is L2-hot).
// ---------------------------------------------------------------------------
__global__ __launch_bounds__(256)
void recnet_readout_kernel(const float* __restrict__ z,     // [B,H]
                           const float* __restrict__ w_out, // [O,H]
                           float* __restrict__ lv,          // [B,O]
                           float* __restrict__ li,          // [B,O]
                           float* __restrict__ out,         // [T,B,O]
                           int t)
{
    const int tid = blockIdx.x * blockDim.x + threadIdx.x;
    if (tid >= B_SZ * O_SZ) return;
    const int b = tid / O_SZ;
    const int o = tid - b * O_SZ;

    const float4* zr = (const float4*)(z     + (size_t)b * H_SZ);
    const float4* wr = (const float4*)(w_out + (size_t)o * H_SZ);
    float y = 0.0f;
    #pragma unroll 4
    for (int k = 0; k < H_SZ / 4; ++k) {
        const float4 a = zr[k];
        const float4 w = wr[k];
        y += a.x * w.x + a.y * w.y + a.z * w.z + a.w * w.w;
    }

    const float lvo = lv[tid];
    const float lio = li[tid];
    const float lvn = lvo + DV * (lio - lvo);   // uses old li (norse LI order)
    li[tid] = lio - DI * lio + y;
    lv[tid] = lvn;
    out[(size_t)t * (B_SZ * O_SZ) + tid] = lvn;
}

// ---------------------------------------------------------------------------
// Workspace zero-init (ws is poisoned 0xAA and never re-poisoned; state must
// be zeroed at the start of EVERY launch for deterministic replays).
// ---------------------------------------------------------------------------
__global__ void recnet_zero_kernel(float* __restrict__ p, int n)
{
    const int i = blockIdx.x * blockDim.x + threadIdx.x;
    if (i < n) p[i] = 0.0f;
}

extern "C" void kernel_launch(void* const* d_in, const int* in_sizes, int n_in,
                              void* d_out, int out_size, void* d_ws, size_t ws_size,
                              hipStream_t stream)
{
    (void)in_sizes; (void)n_in; (void)out_size; (void)ws_size;

    const float* x     = (const float*)d_in[0];   // [T,B,1,28,28] -> [T,B,784]
    const float* w_in  = (const float*)d_in[1];   // [H,F]
    const float* w_rec = (const float*)d_in[2];   // [H,H]
    const float* w_out = (const float*)d_in[3];   // [O,H]
    float*       out   = (float*)d_out;           // [T,B,O]

    // Workspace layout (floats): v, i, z0, z1 : [B,H] each; lv, li : [B,O].
    float* ws   = (float*)d_ws;
    float* v_st = ws;
    float* i_st = v_st + (size_t)B_SZ * H_SZ;
    float* z0   = i_st + (size_t)B_SZ * H_SZ;
    float* z1   = z0   + (size_t)B_SZ * H_SZ;
    float* lv   = z1   + (size_t)B_SZ * H_SZ;
    float* li   = lv   + (size_t)B_SZ * O_SZ;

    const int n_state = 4 * B_SZ * H_SZ + 2 * B_SZ * O_SZ;  // ~4.04 MB
    recnet_zero_kernel<<<(n_state + 255) / 256, 256, 0, stream>>>(ws, n_state);

    float* zp = z0;   // z_{t-1}
    float* zc = z1;   // z_t
    for (int t = 0; t < T_STEPS; ++t) {
        // 1024 wave-tiles (32x32 of 16x16), 4 waves per block.
        recnet_step_kernel<<<256, 128, 0, stream>>>(
            x, w_in, w_rec, zp, zc, v_st, i_st, t);
        recnet_readout_kernel<<<(B_SZ * O_SZ + 255) / 256, 256, 0, stream>>>(
            zc, w_out, lv, li, out, t);
        float* tmp = zp; zp = zc; zc = tmp;   // double-buffer z across steps
    }
}